// GCNClassifier_11441792877137
// MI455X (gfx1250) — compile-verified
//
#include <hip/hip_runtime.h>

// ---------------------------------------------------------------------------
// GCN classifier for MI455X (gfx1250, wave32).
// Bottleneck is the edge scatter (82M f32 atomics/layer) which lives in the
// 192MB L2. GEMMs use native f32 WMMA (V_WMMA_F32_16X16X4_F32) to keep
// reference-exact fp32 numerics (GEMM time is negligible, so no bf16 downcast).
// ---------------------------------------------------------------------------

typedef float v2f __attribute__((ext_vector_type(2)));
typedef float v8f __attribute__((ext_vector_type(8)));

__global__ __launch_bounds__(256) void k_zero(float* __restrict__ p, long long n) {
  long long i = (long long)blockIdx.x * blockDim.x + threadIdx.x;
  if (i < n) p[i] = 0.0f;
}

__global__ __launch_bounds__(256) void k_degree(const int* __restrict__ src,
                                                const int* __restrict__ dst,
                                                float* __restrict__ odeg,
                                                float* __restrict__ ideg, int E) {
  int e = blockIdx.x * blockDim.x + threadIdx.x;
  if (e >= E) return;
  atomicAdd(&odeg[src[e]], 1.0f);
  atomicAdd(&ideg[dst[e]], 1.0f);
}

// deg -> (deg+1)^-0.5 in place (covers out_deg and in_deg contiguously)
__global__ __launch_bounds__(256) void k_rnorm(float* __restrict__ p, int n) {
  int i = blockIdx.x * blockDim.x + threadIdx.x;
  if (i < n) p[i] = rsqrtf(p[i] + 1.0f);
}

// msg = (X[nrows,K] @ W[K,128]) * out_norm[row], via V_WMMA_F32_16X16X4_F32.
// blockDim = (32, 8): 8 waves/block, each wave owns one 16x16 output tile.
// A 16x4 f32: lane(0..15)=row M, vgpr {0,1}=K{0,1}; lanes 16..31 hold K{2,3}.
// B 4x16 f32: lane(0..15)=col N, mirrored K striping.
// C/D 16x16 f32: lane lo = N, vgpr v = M (lo half M=v, hi half M=8+v).
template <int K>
__global__ __launch_bounds__(256) void k_gemm_scale(const float* __restrict__ X,
                                                    const float* __restrict__ W,
                                                    const float* __restrict__ onorm,
                                                    float* __restrict__ out,
                                                    int nrows) {
  const int lane = threadIdx.x;              // 0..31
  const int tile = blockIdx.x * 8 + threadIdx.y;
  const int m0 = (tile >> 3) * 16;           // 8 column tiles (128/16)
  const int n0 = (tile & 7) * 16;
  if (m0 >= nrows) return;
  const int lo = lane & 15;
  const int hi = lane >> 4;                  // 0 or 1

  const float2* __restrict__ Arow = (const float2*)(X + (long long)(m0 + lo) * K);
  const float* __restrict__ Bcol = W + (n0 + lo);

  v8f acc = {};
#pragma unroll
  for (int k0 = 0; k0 < K; k0 += 4) {
    float2 av = Arow[(k0 >> 1) + hi];        // A[m][k0+2*hi .. +1]
    v2f a; a.x = av.x; a.y = av.y;
    v2f b;
    b.x = Bcol[(k0 + 2 * hi) * 128];         // B[k0+2*hi][n]
    b.y = Bcol[(k0 + 2 * hi + 1) * 128];     // B[k0+2*hi+1][n]
    acc = __builtin_amdgcn_wmma_f32_16x16x4_f32(false, a, false, b,
                                                (short)0, acc, false, false);
  }
#pragma unroll
  for (int v = 0; v < 8; ++v) {
    int m = m0 + hi * 8 + v;
    out[(long long)m * 128 + n0 + lo] = acc[v] * onorm[m];
  }
}

// agg[dst] += msg[src], 128 f32 per edge. One wave per edge, float4 per lane.
__global__ __launch_bounds__(256) void k_scatter(const float4* __restrict__ msg,
                                                 const int* __restrict__ src,
                                                 const int* __restrict__ dst,
                                                 float* __restrict__ agg, int E) {
  int e = blockIdx.x * 8 + (threadIdx.x >> 5);
  if (e >= E) return;
  int lane = threadIdx.x & 31;
  int s = src[e], d = dst[e];
  float4 v = msg[(long long)s * 32 + lane];
  float* ap = agg + (long long)d * 128 + lane * 4;
  atomicAdd(ap + 0, v.x);
  atomicAdd(ap + 1, v.y);
  atomicAdd(ap + 2, v.z);
  atomicAdd(ap + 3, v.w);
}

// h = relu((agg + msg) * in_norm + bias)   (msg term = self-loop)
__global__ __launch_bounds__(256) void k_epilogue(const float* __restrict__ agg,
                                                  const float* __restrict__ msg,
                                                  const float* __restrict__ inorm,
                                                  const float* __restrict__ bias,
                                                  float* __restrict__ out, int n) {
  int idx = blockIdx.x * blockDim.x + threadIdx.x;
  if (idx >= n) return;
  int i = idx >> 7, f = idx & 127;
  float v = (agg[idx] + msg[idx]) * inorm[i] + bias[f];
  out[idx] = fmaxf(v, 0.0f);
}

// layer-2 epilogue fused with per-graph sum readout (h2 never materialized)
__global__ __launch_bounds__(256) void k_epilogue_readout(
    const float* __restrict__ agg, const float* __restrict__ msg,
    const float* __restrict__ inorm, const float* __restrict__ bias,
    const int* __restrict__ gid, float* __restrict__ hg, int n) {
  int idx = blockIdx.x * blockDim.x + threadIdx.x;
  if (idx >= n) return;
  int i = idx >> 7, f = idx & 127;
  float v = (agg[idx] + msg[idx]) * inorm[i] + bias[f];
  v = fmaxf(v, 0.0f);
  atomicAdd(&hg[gid[i] * 128 + f], v);
}

__global__ __launch_bounds__(256) void k_count(const int* __restrict__ gid,
                                               float* __restrict__ counts, int N) {
  int i = blockIdx.x * blockDim.x + threadIdx.x;
  if (i < N) atomicAdd(&counts[gid[i]], 1.0f);
}

// out[64,10] = (hg/count) @ Wc + bc  (tiny: plain VALU)
__global__ __launch_bounds__(128) void k_classify(const float* __restrict__ hg,
                                                  const float* __restrict__ counts,
                                                  const float* __restrict__ Wc,
                                                  const float* __restrict__ bc,
                                                  float* __restrict__ out) {
  int idx = blockIdx.x * blockDim.x + threadIdx.x;
  if (idx >= 64 * 10) return;
  int g = idx / 10, c = idx % 10;
  float inv = 1.0f / fmaxf(counts[g], 1.0f);
  float s = bc[c];
#pragma unroll 8
  for (int k = 0; k < 128; ++k) s += hg[g * 128 + k] * inv * Wc[k * 10 + c];
  out[idx] = s;
}

static inline int cdiv(long long a, long long b) { return (int)((a + b - 1) / b); }

extern "C" void kernel_launch(void* const* d_in, const int* in_sizes, int n_in,
                              void* d_out, int out_size, void* d_ws, size_t ws_size,
                              hipStream_t stream) {
  const float* feat = (const float*)d_in[0];  // [N,32]
  const float* W1   = (const float*)d_in[1];  // [32,128]
  const float* b1   = (const float*)d_in[2];  // [128]
  const float* W2   = (const float*)d_in[3];  // [128,128]
  const float* b2   = (const float*)d_in[4];  // [128]
  const float* Wc   = (const float*)d_in[5];  // [128,10]
  const float* bc   = (const float*)d_in[6];  // [10]
  const int*   src  = (const int*)d_in[7];    // [E]
  const int*   dst  = (const int*)d_in[8];    // [E]
  const int*   gid  = (const int*)d_in[9];    // [N], sorted
  float* out = (float*)d_out;                 // [64,10]

  const int N = in_sizes[0] / 32;
  const int E = in_sizes[7];
  const long long NH = (long long)N * 128;

  // workspace layout (floats): out_norm[N] | in_norm[N] | bufA[N*128] |
  //                            bufB[N*128] | hg[64*128] | counts[64]
  float* ws     = (float*)d_ws;
  float* onorm  = ws;
  float* inorm  = ws + N;
  float* bufA   = ws + 2LL * N;
  float* bufB   = bufA + NH;
  float* hg     = bufB + NH;
  float* counts = hg + 64 * 128;

  // degree norms: (deg+1)^-0.5 for out(src) and in(dst)
  k_zero<<<cdiv(2LL * N, 256), 256, 0, stream>>>(onorm, 2LL * N);
  k_degree<<<cdiv(E, 256), 256, 0, stream>>>(src, dst, onorm, inorm, E);
  k_rnorm<<<cdiv(2LL * N, 256), 256, 0, stream>>>(onorm, 2 * N);

  dim3 wblk(32, 8);
  int gemm_blocks = cdiv((long long)(N / 16) * 8, 8);  // 8 waves/block

  // ---- layer 1: msg1=A, agg1=B, h1 -> A (in-place elementwise) ----
  k_gemm_scale<32><<<gemm_blocks, wblk, 0, stream>>>(feat, W1, onorm, bufA, N);
  k_zero<<<cdiv(NH, 256), 256, 0, stream>>>(bufB, NH);
  k_scatter<<<cdiv(E, 8), 256, 0, stream>>>((const float4*)bufA, src, dst, bufB, E);
  k_epilogue<<<cdiv(NH, 256), 256, 0, stream>>>(bufB, bufA, inorm, b1, bufA, (int)NH);

  // ---- layer 2: msg2=B (reads A), agg2=A (zeroed after GEMM2), fused readout ----
  k_gemm_scale<128><<<gemm_blocks, wblk, 0, stream>>>(bufA, W2, onorm, bufB, N);
  k_zero<<<cdiv(NH, 256), 256, 0, stream>>>(bufA, NH);
  k_scatter<<<cdiv(E, 8), 256, 0, stream>>>((const float4*)bufB, src, dst, bufA, E);

  k_zero<<<cdiv(64 * 128 + 64, 256), 256, 0, stream>>>(hg, 64 * 128 + 64);
  k_count<<<cdiv(N, 256), 256, 0, stream>>>(gid, counts, N);
  k_epilogue_readout<<<cdiv(NH, 256), 256, 0, stream>>>(bufA, bufB, inorm, b2, gid, hg, (int)NH);

  k_classify<<<cdiv(640, 128), 128, 0, stream>>>(hg, counts, Wc, bc, out);
}